// PoseLoss_33311766348301
// MI455X (gfx1250) — compile-verified
//
#include <hip/hip_runtime.h>
#include <hip/hip_bf16.h>
#include <math.h>

typedef __attribute__((ext_vector_type(2))) float v2f;
typedef __attribute__((ext_vector_type(8))) float v8f;

#define NBLK 2048
#define NTHR 256

// ---------------------------------------------------------------------------
// Wave32 full reduction using V_WMMA_F32_16X16X4_F32 (f32 -> f32, no precision
// loss). B matrix = all ones (layout-independent), so D[i][j] = rowsum(A,i).
// With each lane's partial placed in A VGPR0 (K=0 for lanes 0-15, K=2 for
// lanes 16-31 per the ISA 16x4 f32 A layout) and VGPR1 = 0:
//   rowsum_l = p_l + p_{l+16}
// C/D layout: lane j<16 holds rows 0..7, lane j>=16 holds rows 8..15, so
// summing the 8 D VGPRs and adding the xor-16 partner yields the 32-lane sum
// in every lane. Requires EXEC == all ones (true at call sites).
// ---------------------------------------------------------------------------
__device__ __forceinline__ float wave_sum_wmma(float p) {
    v2f a; a[0] = p;    a[1] = 0.0f;
    v2f b; b[0] = 1.0f; b[1] = 1.0f;
    v8f c = {};
    c = __builtin_amdgcn_wmma_f32_16x16x4_f32(
        /*neg_a=*/false, a, /*neg_b=*/false, b,
        /*c_mod=*/(short)0, c, /*reuse_a=*/false, /*reuse_b=*/false);
    float s = ((c[0] + c[1]) + (c[2] + c[3])) + ((c[4] + c[5]) + (c[6] + c[7]));
    s += __shfl_xor(s, 16, 32);
    return s;
}

__device__ __forceinline__ void euler_R(float yaw, float pitch, float roll,
                                        float R[9]) {
    float cy, sy, cp, sp, cr, sr;
    sincosf(yaw,   &sy, &cy);
    sincosf(pitch, &sp, &cp);
    sincosf(roll,  &sr, &cr);
    R[0] = cy * cp; R[1] = cy * sp * sr - sy * cr; R[2] = cy * sp * cr + sy * sr;
    R[3] = sy * cp; R[4] = sy * sp * sr + cy * cr; R[5] = sy * sp * cr - cy * sr;
    R[6] = -sp;     R[7] = cp * sr;                R[8] = cp * cr;
}

// ---------------------------------------------------------------------------
// Kernel 1: grid-stride over poses (b64 loads; 24*i is always 8B aligned) and
// over class data (b128 float4 loads). Per-thread fp32 accumulators, WMMA wave
// reduction, LDS cross-wave reduction, one partial pair per block to ws.
// ---------------------------------------------------------------------------
__global__ __launch_bounds__(NTHR) void pose_class_partials(
    const float* __restrict__ pred_pose, const float* __restrict__ target_pose,
    const float4* __restrict__ pred_class, const float4* __restrict__ target_class,
    int B, int NC4, float* __restrict__ ws) {
    const float L = 0.05f;
    int tid = blockIdx.x * NTHR + threadIdx.x;
    int stride = gridDim.x * NTHR;

    float pose_acc = 0.0f;
    for (int i = tid; i < B; i += stride) {
        const float2* pp = reinterpret_cast<const float2*>(pred_pose) + 3 * (size_t)i;
        const float2* tp = reinterpret_cast<const float2*>(target_pose) + 3 * (size_t)i;
        float2 p0 = pp[0], p1 = pp[1], p2 = pp[2];
        float2 q0 = tp[0], q1 = tp[1], q2 = tp[2];

        float Rp[9], Rg[9];
        euler_R(p0.x, p0.y, p1.x, Rp);
        euler_R(q0.x, q0.y, q1.x, Rg);

        // Columns of L*(Rp - Rg) and translation difference
        float ax = L * (Rp[0] - Rg[0]), ay = L * (Rp[3] - Rg[3]), az = L * (Rp[6] - Rg[6]);
        float bx = L * (Rp[1] - Rg[1]), by = L * (Rp[4] - Rg[4]), bz = L * (Rp[7] - Rg[7]);
        float cx = L * (Rp[2] - Rg[2]), cy = L * (Rp[5] - Rg[5]), cz = L * (Rp[8] - Rg[8]);
        float dx = p1.y - q1.y, dy = p2.x - q2.x, dz = p2.y - q2.y;

        float s8 = 0.0f;
#pragma unroll
        for (int n = 0; n < 8; ++n) {
            float sx = (n & 4) ? 1.0f : -1.0f;
            float sy = (n & 2) ? 1.0f : -1.0f;
            float sz = (n & 1) ? 1.0f : -1.0f;
            float vx = fmaf(sx, ax, fmaf(sy, bx, fmaf(sz, cx, dx)));
            float vy = fmaf(sx, ay, fmaf(sy, by, fmaf(sz, cy, dy)));
            float vz = fmaf(sx, az, fmaf(sy, bz, fmaf(sz, cz, dz)));
            s8 += sqrtf(fmaf(vx, vx, fmaf(vy, vy, vz * vz)));
        }
        pose_acc += 0.125f * s8;
    }

    float class_acc = 0.0f;
    for (int i = tid; i < NC4; i += stride) {
        float4 p = pred_class[i];
        float4 q = target_class[i];
        float e0 = p.x - q.x, e1 = p.y - q.y, e2 = p.z - q.z, e3 = p.w - q.w;
        class_acc += fmaf(e0, e0, fmaf(e1, e1, fmaf(e2, e2, e3 * e3)));
    }

    // Wave reduction via WMMA, then cross-wave via LDS
    float wp = wave_sum_wmma(pose_acc);
    float wc = wave_sum_wmma(class_acc);

    __shared__ float red_p[NTHR / 32];
    __shared__ float red_c[NTHR / 32];
    int wave = threadIdx.x >> 5;
    int lane = threadIdx.x & 31;
    if (lane == 0) { red_p[wave] = wp; red_c[wave] = wc; }
    __syncthreads();
    if (threadIdx.x == 0) {
        float tp = 0.0f, tc = 0.0f;
#pragma unroll
        for (int w = 0; w < NTHR / 32; ++w) { tp += red_p[w]; tc += red_c[w]; }
        ws[2 * blockIdx.x + 0] = tp;
        ws[2 * blockIdx.x + 1] = tc;
    }
}

// ---------------------------------------------------------------------------
// Kernel 2: single block folds the NBLK partial pairs (deterministic tree:
// strided per-thread sums -> WMMA wave reduction -> LDS) and emits the three
// scalars in reference return order: (total, loss_pose, class_loss).
// ---------------------------------------------------------------------------
__global__ __launch_bounds__(NTHR) void final_reduce(
    const float* __restrict__ ws, int nblk, float inv_pose_n, float inv_class_n,
    float* __restrict__ out) {
    float pose_acc = 0.0f, class_acc = 0.0f;
    for (int i = threadIdx.x; i < nblk; i += NTHR) {
        pose_acc  += ws[2 * i + 0];
        class_acc += ws[2 * i + 1];
    }

    float wp = wave_sum_wmma(pose_acc);
    float wc = wave_sum_wmma(class_acc);

    __shared__ float red_p[NTHR / 32];
    __shared__ float red_c[NTHR / 32];
    int wave = threadIdx.x >> 5;
    int lane = threadIdx.x & 31;
    if (lane == 0) { red_p[wave] = wp; red_c[wave] = wc; }
    __syncthreads();
    if (threadIdx.x == 0) {
        float tp = 0.0f, tc = 0.0f;
#pragma unroll
        for (int w = 0; w < NTHR / 32; ++w) { tp += red_p[w]; tc += red_c[w]; }
        float loss_pose  = tp * inv_pose_n;
        float class_loss = tc * inv_class_n;
        out[0] = loss_pose + class_loss;
        out[1] = loss_pose;
        out[2] = class_loss;
    }
}

extern "C" void kernel_launch(void* const* d_in, const int* in_sizes, int n_in,
                              void* d_out, int out_size, void* d_ws, size_t ws_size,
                              hipStream_t stream) {
    const float*  pred_pose    = (const float*)d_in[0];
    const float4* pred_class   = (const float4*)d_in[1];
    const float*  target_pose  = (const float*)d_in[2];
    const float4* target_class = (const float4*)d_in[3];

    int B   = in_sizes[0] / 6;   // number of poses
    int NC  = in_sizes[1];       // total class elements (B*C)
    int NC4 = NC / 4;

    float* ws  = (float*)d_ws;   // NBLK * 2 floats = 16 KB
    float* out = (float*)d_out;

    pose_class_partials<<<NBLK, NTHR, 0, stream>>>(
        pred_pose, target_pose, pred_class, target_class, B, NC4, ws);
    final_reduce<<<1, NTHR, 0, stream>>>(
        ws, NBLK, 1.0f / (float)B, 1.0f / (float)NC, out);
}